// ChebConv_26250840113269
// MI455X (gfx1250) — compile-verified
//
#include <hip/hip_runtime.h>

// ---------------------------------------------------------------------------
// ChebConv (K=6) for gfx1250 / MI455X, wave32.
//   - SpMV (L_hat matvec) via float4 gather + hardware f32 global atomics.
//   - Dense accumulation via V_WMMA_F32_16X16X4_F32 (exact fp32 matrix core),
//     with W staged in LDS as K-pair float2 (single ds_load_b64 per B frag,
//     XOR-16 swizzle for conflict-free half-wave access).
// ---------------------------------------------------------------------------

typedef float v2f __attribute__((ext_vector_type(2)));
typedef float v8f __attribute__((ext_vector_type(8)));

#define NNODES 50000
#define NEDGES 800000
#define CH     128
#define KCHEB  6
#define NTILES (NNODES / 16)   // 3125, exact

__device__ __forceinline__ float atomicAddF(float* p, float v) {
  return __hip_atomic_fetch_add(p, v, __ATOMIC_RELAXED, __HIP_MEMORY_SCOPE_AGENT);
}

// Monotonic float->uint key (total order preserved under unsigned compare).
__device__ __forceinline__ unsigned encf(float f) {
  unsigned u = __float_as_uint(f);
  return (u & 0x80000000u) ? ~u : (u | 0x80000000u);
}
__device__ __forceinline__ float decf(unsigned k) {
  return (k & 0x80000000u) ? __uint_as_float(k & 0x7fffffffu)
                           : __uint_as_float(~k);
}

// --------------------------- setup kernels ---------------------------------

__global__ void k_init_deg(float* __restrict__ deg, unsigned* __restrict__ maxkey) {
  int i = blockIdx.x * blockDim.x + threadIdx.x;
  if (i < NNODES) deg[i] = 0.0f;
  if (i == 0) *maxkey = 0u;  // encf(most negative)
}

__global__ void k_deg_scatter(const float* __restrict__ edges,
                              const int* __restrict__ senders,
                              float* __restrict__ deg) {
  int e = blockIdx.x * blockDim.x + threadIdx.x;
  if (e < NEDGES) atomicAddF(&deg[senders[e]], edges[e]);
}

__global__ void k_reduce_max(const float* __restrict__ deg,
                             const float* __restrict__ edges,
                             unsigned* __restrict__ maxkey) {
  float m = -3.4e38f;
  for (int i = blockIdx.x * blockDim.x + threadIdx.x; i < NEDGES;
       i += gridDim.x * blockDim.x) {
    float v = -edges[i];
    if (i < NNODES) v = fmaxf(v, deg[i]);
    m = fmaxf(m, v);
  }
  for (int off = 16; off > 0; off >>= 1)
    m = fmaxf(m, __shfl_xor(m, off, 32));
  if ((threadIdx.x & 31) == 0) atomicMax(maxkey, encf(m));
}

__global__ void k_scale(const unsigned* __restrict__ maxkey, float* __restrict__ scalep) {
  // lambda_max = 2*m ; scale = 2/lambda_max = 1/m
  *scalep = 1.0f / decf(*maxkey);
}

// out[i][c] = bias[c] + sum_k b_dense[k][c]
__global__ void k_out_init(const float* __restrict__ b_dense,
                           const float* __restrict__ bias,
                           float* __restrict__ out) {
  int i = blockIdx.x * blockDim.x + threadIdx.x;
  if (i >= NNODES * CH) return;
  int c = i & (CH - 1);
  float s = bias[c];
#pragma unroll
  for (int k = 0; k < KCHEB; ++k) s += b_dense[k * CH + c];
  out[i] = s;
}

// ------------------------ Chebyshev recurrence -----------------------------

// y[i][c] = coef*scale*deg[i]*x1[i][c] - (x0 ? x0[i][c] : 0)
// NOTE: y may alias x0 (elementwise read-then-write), so no __restrict__ there.
__global__ void k_cheb_init(float* y, const float* __restrict__ x1, const float* x0,
                            const float* __restrict__ deg,
                            const float* __restrict__ scalep, float coef) {
  int i = blockIdx.x * blockDim.x + threadIdx.x;
  if (i >= NNODES * CH) return;
  int node = i >> 7;  // CH == 128
  float v = coef * (*scalep) * deg[node] * x1[i];
  if (x0) v -= x0[i];
  y[i] = v;
}

// y[s][*] -= coef*scale*e * x1[r][*]   (one wave per edge, lane = 4 channels)
__global__ void k_cheb_scatter(float* __restrict__ y, const float* __restrict__ x1,
                               const float* __restrict__ edges,
                               const int* __restrict__ senders,
                               const int* __restrict__ receivers,
                               const float* __restrict__ scalep, float coef) {
  int t = blockIdx.x * blockDim.x + threadIdx.x;
  int e = t >> 5;
  int lane = t & 31;
  if (e >= NEDGES) return;
  int s = senders[e];
  int r = receivers[e];
  float w = -coef * (*scalep) * edges[e];
  const float4 xv = *(const float4*)(x1 + (size_t)r * CH + lane * 4);
  float* yp = y + (size_t)s * CH + lane * 4;
  atomicAddF(yp + 0, w * xv.x);
  atomicAddF(yp + 1, w * xv.y);
  atomicAddF(yp + 2, w * xv.z);
  atomicAddF(yp + 3, w * xv.w);
}

// ----------------------------- WMMA GEMM -----------------------------------
// out[50000x128] += X[50000x128] @ Wk[128x128], exact fp32 via
// V_WMMA_F32_16X16X4_F32. One wave computes a 16(M) x 128(N) strip.
//
// A frag (16x4 f32): lane L<16 holds X[m0+L][k..k+1], lane L>=16 holds k+2..k+3.
// B frag (4x16 f32): lane L<16 holds Wk[k][n16+L],Wk[k+1][n16+L]; upper lanes k+2,k+3.
// C/D (16x16 f32): vgpr j = row j (lanes 0-15) / row j+8 (lanes 16-31), col = lane%16.
//
// LDS image of Wk: float2 per (pair p = k/2, col): (Wk[2p][col], Wk[2p+1][col]),
// stored at column (col ^ ((p&1)<<4)). One B frag = one aligned ds_load_b64.
// Within a WMMA step the two half-waves use pairs p and p+1 (opposite parity),
// so the XOR-16 swizzle puts them in disjoint bank halves -> conflict-free.

__global__ __launch_bounds__(128) void k_gemm_wmma(const float* __restrict__ X,
                                                   const float* __restrict__ Wk,
                                                   float* __restrict__ out,
                                                   int nTiles) {
  __shared__ float2 lw2[(CH / 2) * CH];  // 64 KB
  for (int i = threadIdx.x; i < (CH / 2) * CH; i += blockDim.x) {
    int p = i >> 7;        // K pair
    int col = i & (CH - 1);
    float2 w2;
    w2.x = Wk[(2 * p) * CH + col];
    w2.y = Wk[(2 * p + 1) * CH + col];
    lw2[p * CH + (col ^ ((p & 1) << 4))] = w2;
  }
  __syncthreads();

  int wave = threadIdx.x >> 5;
  int lane = threadIdx.x & 31;
  int mt = blockIdx.x * 4 + wave;
  if (mt >= nTiles) return;  // wave-uniform: EXEC stays all-ones for WMMA

  int l16 = lane & 15;
  int kh = (lane >> 4) * 2;  // 0 or 2
  const float* xrow = X + (size_t)(mt * 16 + l16) * CH;

  v8f acc[8];
#pragma unroll
  for (int n = 0; n < 8; ++n) acc[n] = (v8f)(0.0f);

#pragma unroll 4
  for (int kk = 0; kk < CH; kk += 4) {
    float2 a2 = *(const float2*)(xrow + kk + kh);
    v2f av;
    av.x = a2.x;
    av.y = a2.y;
    int p = (kk + kh) >> 1;               // this half-wave's K pair
    const float2* brow = lw2 + p * CH;
    int sw = (p & 1) << 4;
#pragma unroll
    for (int n = 0; n < 8; ++n) {
      int col = (n * 16 + l16) ^ sw;
      float2 b2 = brow[col];              // single ds_load_b64, even-aligned
      v2f bv;
      bv.x = b2.x;
      bv.y = b2.y;
      acc[n] = __builtin_amdgcn_wmma_f32_16x16x4_f32(
          /*neg_a=*/false, av, /*neg_b=*/false, bv,
          /*c_mod=*/(short)0, acc[n], /*reuse_a=*/false, /*reuse_b=*/false);
    }
  }

  int rbase = mt * 16 + (lane >> 4) * 8;
#pragma unroll
  for (int n = 0; n < 8; ++n) {
#pragma unroll
    for (int j = 0; j < 8; ++j) {
      float* p = out + (size_t)(rbase + j) * CH + n * 16 + l16;
      *p += acc[n][j];
    }
  }
}

// ------------------------------ launcher -----------------------------------

extern "C" void kernel_launch(void* const* d_in, const int* in_sizes, int n_in,
                              void* d_out, int out_size, void* d_ws, size_t ws_size,
                              hipStream_t stream) {
  (void)in_sizes; (void)n_in; (void)out_size; (void)ws_size;

  const float* nodes     = (const float*)d_in[0];  // 50000*128
  const float* edges     = (const float*)d_in[1];  // 800000
  const int*   senders   = (const int*)d_in[2];    // 800000
  const int*   receivers = (const int*)d_in[3];    // 800000
  const float* W         = (const float*)d_in[4];  // 6*128*128
  const float* b_dense   = (const float*)d_in[5];  // 6*128
  const float* bias      = (const float*)d_in[6];  // 128
  float* out = (float*)d_out;

  // Workspace layout (needs ~52.3 MB)
  float*    deg    = (float*)d_ws;                        // 50000 f32
  unsigned* maxkey = (unsigned*)((char*)d_ws + 204800);   // 1 u32
  float*    scalep = (float*)((char*)d_ws + 204804);      // 1 f32
  float* B0 = (float*)((char*)d_ws + (1 << 20));          // 50000*128 f32
  float* B1 = B0 + (size_t)NNODES * CH;                   // 50000*128 f32

  const int ELEMS = NNODES * CH;

  // 1) degree + lambda_max + scale
  k_init_deg<<<(NNODES + 255) / 256, 256, 0, stream>>>(deg, maxkey);
  k_deg_scatter<<<(NEDGES + 255) / 256, 256, 0, stream>>>(edges, senders, deg);
  k_reduce_max<<<512, 256, 0, stream>>>(deg, edges, maxkey);
  k_scale<<<1, 1, 0, stream>>>(maxkey, scalep);

  // 2) out = bias + sum_k b_dense[k]
  k_out_init<<<(ELEMS + 255) / 256, 256, 0, stream>>>(b_dense, bias, out);

  dim3 gGemm((NTILES + 3) / 4), bGemm(128);
  dim3 gElem((ELEMS + 255) / 256), bElem(256);
  dim3 gScat((NEDGES * 32 + 255) / 256), bScat(256);

  // k = 0: out += nodes @ W0
  k_gemm_wmma<<<gGemm, bGemm, 0, stream>>>(nodes, W, out, NTILES);

  // k = 1: Tx1 = L_hat(nodes); out += Tx1 @ W1
  k_cheb_init<<<gElem, bElem, 0, stream>>>(B0, nodes, nullptr, deg, scalep, 1.0f);
  k_cheb_scatter<<<gScat, bScat, 0, stream>>>(B0, nodes, edges, senders, receivers,
                                              scalep, 1.0f);
  k_gemm_wmma<<<gGemm, bGemm, 0, stream>>>(B0, W + 1 * CH * CH, out, NTILES);

  // k = 2..5: Tx2 = 2*L_hat(Tx1) - Tx0 ; out += Tx2 @ Wk
  const float* tx0 = nodes;
  float* tx1 = B0;
  for (int k = 2; k < KCHEB; ++k) {
    // Tx2 may reuse the Tx0 buffer (elementwise in init; scatter touches only
    // tx2/tx1). First iteration needs the fresh buffer B1 since tx0 == nodes.
    float* tx2 = (k == 2) ? B1 : (float*)tx0;
    k_cheb_init<<<gElem, bElem, 0, stream>>>(tx2, tx1, tx0, deg, scalep, 2.0f);
    k_cheb_scatter<<<gScat, bScat, 0, stream>>>(tx2, tx1, edges, senders, receivers,
                                                scalep, 2.0f);
    k_gemm_wmma<<<gGemm, bGemm, 0, stream>>>(tx2, W + (size_t)k * CH * CH, out, NTILES);
    tx0 = tx1;
    tx1 = tx2;
  }
}